// TPlanesEnc_59450937311384
// MI455X (gfx1250) — compile-verified
//
#include <hip/hip_runtime.h>
#include <stdint.h>

typedef float v2f __attribute__((ext_vector_type(2)));
typedef float v4f __attribute__((ext_vector_type(4)));

#define S_DIM   512
#define HW      (512 * 512)
#define C_DIM   64
#define NPLANES 3

#if defined(__has_builtin)
#if __has_builtin(__builtin_amdgcn_tensor_load_to_lds)
#define HAVE_TDM 1
#endif
#endif

#if defined(HAVE_TDM)
typedef unsigned int u32x4 __attribute__((ext_vector_type(4)));
typedef int          i32x4 __attribute__((ext_vector_type(4)));
typedef int          i32x8 __attribute__((ext_vector_type(8)));
#if __has_include(<hip/amd_detail/amd_gfx1250_TDM.h>)
#define TDM_SIX_ARGS 1   // amdgpu-toolchain / clang-23: 6-arg builtin
#endif
#endif

// ---------------------------------------------------------------------------
// Pass 1: re-tile (p, C, H, W) -> (p, H, W, C) so the 64 channels of a texel
// are one contiguous 256B block.
// CDNA5 path: the load phase (64 rows x 256B, row stride HW*4B, into a
// bank-conflict-free padded LDS tile) is a textbook Tensor Data Mover job:
// one TENSOR_LOAD_TO_LDS descriptor per block with pad_interval=64 DWORDs /
// pad_amount=1 DWORD reproduces the tile[64][65] layout in hardware.
// Fallback: manual NT b128 loads + ds_store.
// ---------------------------------------------------------------------------
__global__ __launch_bounds__(256) void tplanes_transpose_kernel(
    const float* __restrict__ tp, float* __restrict__ tpT)
{
    __shared__ float tile[64][65];              // +1 pad: conflict-free column reads
    const int b   = blockIdx.x;                 // 3 planes * 4096 hw-tiles
    const int p   = b >> 12;
    const int hw0 = (b & 4095) << 6;
    const int t   = threadIdx.x;

    const float* src = tp + (size_t)p * C_DIM * HW;

#if defined(HAVE_TDM)
    if (t < 32) {  // wave 0 issues one DMA for the whole block (EXEC-independent op)
        const uint64_t gaddr    = (uint64_t)(uintptr_t)(src + hw0); // tile start (c=0,hw0)
        const uint32_t lds_addr = (uint32_t)(uintptr_t)(void*)&tile[0][0];

        u32x4 g0;
        g0[0] = 1u;                                            // count=1 (valid user D#)
        g0[1] = lds_addr;                                      // lds_addr (bytes)
        g0[2] = (uint32_t)(gaddr & 0xFFFFFFFFu);               // global_addr[31:0]
        g0[3] = (uint32_t)((gaddr >> 32) & 0x1FFFFFFu)         // global_addr[56:32]
              | (2u << 30);                                    // type=2 ("image")

        i32x8 g1;
        g1[0] = (int)((2u << 16)      // data_size = 4 bytes
                    | (1u << 20)      // pad_enable
                    | (5u << 22)      // pad_interval: 64 DWORDs
                    | (0u << 25));    // pad_amount: 1 DWORD
        g1[1] = (int)(((unsigned)HW & 0xFFFFu) << 16);         // abar=0 | tensor_dim0 lo
        g1[2] = (int)((((unsigned)HW >> 16) & 0xFFFFu)         // tensor_dim0 hi
                    | (64u << 16));                            // tensor_dim1 lo = 64
        g1[3] = (int)(64u << 16);                              // tensor_dim1 hi=0 | tile_dim0=64
        g1[4] = (int)(64u);                                    // tile_dim1=64 | tile_dim2=0
        g1[5] = (int)((unsigned)HW);                           // tensor_dim0_stride[31:0]
        g1[6] = 0;                                             // stride0 hi | stride1 lo
        g1[7] = 0;                                             // stride1 hi

        i32x4 z4 = {0, 0, 0, 0};                               // groups 2/3: unused (2D tile)
#if defined(TDM_SIX_ARGS)
        i32x8 z8 = {0, 0, 0, 0, 0, 0, 0, 0};
        __builtin_amdgcn_tensor_load_to_lds(g0, g1, z4, z4, z8, 0);
#else
        __builtin_amdgcn_tensor_load_to_lds(g0, g1, z4, z4, 0);
#endif
        __builtin_amdgcn_s_wait_tensorcnt(0);
    }
    __syncthreads();
#else
    {
        const int hwl = (t & 15) * 4;
        const int c0  = t >> 4;
#pragma unroll
        for (int i = 0; i < 4; ++i) {
            const int c = c0 + i * 16;
            v4f v = __builtin_nontemporal_load(
                (const v4f*)(src + (size_t)c * HW + hw0 + hwl));
            tile[c][hwl + 0] = v.x;
            tile[c][hwl + 1] = v.y;
            tile[c][hwl + 2] = v.z;
            tile[c][hwl + 3] = v.w;
        }
    }
    __syncthreads();
#endif

    float* dst = tpT + ((size_t)p * HW + hw0) * C_DIM;
    const int c4 = (t & 15) * 4;
    const int h0 = t >> 4;
#pragma unroll
    for (int i = 0; i < 4; ++i) {
        const int h = h0 + i * 16;
        v4f v;
        v.x = tile[c4 + 0][h];
        v.y = tile[c4 + 1][h];
        v.z = tile[c4 + 2][h];
        v.w = tile[c4 + 3][h];
        *(v4f*)(dst + (size_t)h * C_DIM + c4) = v;
    }
}

// ---------------------------------------------------------------------------
// Pass 2: gather. One point per wave32; lane owns channels {2*lane, 2*lane+1}.
// Per plane: 4 texels x float2 per lane = 4 fully-coalesced 256B global_load_b64
// bursts per wave from the L2-resident channel-last planes. Border handled
// branch-free via clamped indices + zero-masked weights. Output (403MB stream)
// written non-temporally so it cannot evict the 192MB plane set from L2.
// ---------------------------------------------------------------------------
__global__ __launch_bounds__(256) void tplanes_gather_cl_kernel(
    const float* __restrict__ coords, const float* __restrict__ tpT,
    float* __restrict__ out, int num_points)
{
    const int lane = threadIdx.x & 31;
    const int g    = blockIdx.x * 8 + (threadIdx.x >> 5);
    if (g >= num_points) return;

    const float cx = coords[3 * g + 0];
    const float cy = coords[3 * g + 1];
    const float cz = coords[3 * g + 2];

    // inv(TRI_PLANE_AXES)[p] are coordinate swaps: p0 -> (x,y), p1 -> (x,z), p2 -> (z,y)
    const float gxs[3] = {cx, cx, cz};
    const float gys[3] = {cy, cz, cy};

    float* op = out + (size_t)g * 192 + 2 * lane;
    const unsigned laneoff = 2u * (unsigned)lane;

#pragma unroll
    for (int p = 0; p < NPLANES; ++p) {
        // match reference op order exactly: ((g+1)*W - 1) * 0.5
        const float ix = ((gxs[p] + 1.0f) * 512.0f - 1.0f) * 0.5f;
        const float iy = ((gys[p] + 1.0f) * 512.0f - 1.0f) * 0.5f;
        const float x0f = floorf(ix);
        const float y0f = floorf(iy);
        const float wx = ix - x0f;
        const float wy = iy - y0f;
        const int x0 = (int)x0f, y0 = (int)y0f;
        const int x1 = x0 + 1,  y1 = y0 + 1;

        float w00 = (1.0f - wy) * (1.0f - wx);
        float w01 = (1.0f - wy) * wx;
        float w10 = wy * (1.0f - wx);
        float w11 = wy * wx;
        const bool vx0 = (unsigned)x0 < 512u, vx1 = (unsigned)x1 < 512u;
        const bool vy0 = (unsigned)y0 < 512u, vy1 = (unsigned)y1 < 512u;
        w00 = (vx0 & vy0) ? w00 : 0.0f;
        w01 = (vx1 & vy0) ? w01 : 0.0f;
        w10 = (vx0 & vy1) ? w10 : 0.0f;
        w11 = (vx1 & vy1) ? w11 : 0.0f;
        const int xc0 = min(max(x0, 0), 511);
        const int xc1 = min(x1, 511);
        const int yc0 = min(max(y0, 0), 511);
        const int yc1 = min(y1, 511);

        const float* plane = tpT + (size_t)p * HW * C_DIM;
        const unsigned o00 = (unsigned)(yc0 * 512 + xc0) * 64u + laneoff;
        const unsigned o01 = (unsigned)(yc0 * 512 + xc1) * 64u + laneoff;
        const unsigned o10 = (unsigned)(yc1 * 512 + xc0) * 64u + laneoff;
        const unsigned o11 = (unsigned)(yc1 * 512 + xc1) * 64u + laneoff;
        const v2f v00 = *(const v2f*)(plane + o00);   // 8B-aligned by construction
        const v2f v01 = *(const v2f*)(plane + o01);
        const v2f v10 = *(const v2f*)(plane + o10);
        const v2f v11 = *(const v2f*)(plane + o11);

        float r0 = v00.x * w00;
        r0 = fmaf(v01.x, w01, r0);
        r0 = fmaf(v10.x, w10, r0);
        r0 = fmaf(v11.x, w11, r0);
        float r1 = v00.y * w00;
        r1 = fmaf(v01.y, w01, r1);
        r1 = fmaf(v10.y, w10, r1);
        r1 = fmaf(v11.y, w11, r1);

        v2f res; res.x = r0; res.y = r1;
        __builtin_nontemporal_store(res, (v2f*)(op + p * 64));
    }
}

// ---------------------------------------------------------------------------
// Fallback (ws too small): direct gather on the original channel-major layout.
// ---------------------------------------------------------------------------
__global__ __launch_bounds__(256) void tplanes_gather_direct_kernel(
    const float* __restrict__ coords, const float* __restrict__ tp,
    float* __restrict__ out, int num_points)
{
    const int lane = threadIdx.x & 31;
    const int g    = blockIdx.x * 8 + (threadIdx.x >> 5);
    if (g >= num_points) return;

    const float cx = coords[3 * g + 0];
    const float cy = coords[3 * g + 1];
    const float cz = coords[3 * g + 2];
    const float gxs[3] = {cx, cx, cz};
    const float gys[3] = {cy, cz, cy};

    float* op = out + (size_t)g * 192 + lane;

#pragma unroll
    for (int p = 0; p < NPLANES; ++p) {
        const float ix = ((gxs[p] + 1.0f) * 512.0f - 1.0f) * 0.5f;
        const float iy = ((gys[p] + 1.0f) * 512.0f - 1.0f) * 0.5f;
        const float x0f = floorf(ix);
        const float y0f = floorf(iy);
        const float wx = ix - x0f;
        const float wy = iy - y0f;
        const int x0 = (int)x0f, y0 = (int)y0f;
        const int x1 = x0 + 1,  y1 = y0 + 1;

        float w00 = (1.0f - wy) * (1.0f - wx);
        float w01 = (1.0f - wy) * wx;
        float w10 = wy * (1.0f - wx);
        float w11 = wy * wx;
        const bool vx0 = (unsigned)x0 < 512u, vx1 = (unsigned)x1 < 512u;
        const bool vy0 = (unsigned)y0 < 512u, vy1 = (unsigned)y1 < 512u;
        w00 = (vx0 & vy0) ? w00 : 0.0f;
        w01 = (vx1 & vy0) ? w01 : 0.0f;
        w10 = (vx0 & vy1) ? w10 : 0.0f;
        w11 = (vx1 & vy1) ? w11 : 0.0f;
        const unsigned o00 = (unsigned)(min(max(y0, 0), 511) * 512 + min(max(x0, 0), 511));
        const unsigned o01 = (unsigned)(min(max(y0, 0), 511) * 512 + min(x1, 511));
        const unsigned o10 = (unsigned)(min(y1, 511) * 512 + min(max(x0, 0), 511));
        const unsigned o11 = (unsigned)(min(y1, 511) * 512 + min(x1, 511));

        const float* pc0 = tp + ((size_t)(p * C_DIM) + lane) * HW;
        const float* pc1 = pc0 + (size_t)32 * HW;

        float r0 = pc0[o00] * w00;
        r0 = fmaf(pc0[o01], w01, r0);
        r0 = fmaf(pc0[o10], w10, r0);
        r0 = fmaf(pc0[o11], w11, r0);
        float r1 = pc1[o00] * w00;
        r1 = fmaf(pc1[o01], w01, r1);
        r1 = fmaf(pc1[o10], w10, r1);
        r1 = fmaf(pc1[o11], w11, r1);

        __builtin_nontemporal_store(r0, op + p * 64);
        __builtin_nontemporal_store(r1, op + p * 64 + 32);
    }
}

extern "C" void kernel_launch(void* const* d_in, const int* in_sizes, int n_in,
                              void* d_out, int out_size, void* d_ws, size_t ws_size,
                              hipStream_t stream) {
    const float* coords = (const float*)d_in[0];   // (N*M, 3) f32
    const float* tp     = (const float*)d_in[1];   // (3, 64, 512, 512) f32
    float* out          = (float*)d_out;           // (N*M, 192) f32

    const int num_points    = in_sizes[0] / 3;
    const int gather_blocks = (num_points + 7) / 8;          // 8 waves/block, 1 point/wave
    const size_t needed     = (size_t)NPLANES * C_DIM * HW * sizeof(float); // 192 MB

    if (ws_size >= needed) {
        float* tpT = (float*)d_ws;
        tplanes_transpose_kernel<<<NPLANES * (HW / 64), 256, 0, stream>>>(tp, tpT);
        tplanes_gather_cl_kernel<<<gather_blocks, 256, 0, stream>>>(coords, tpT, out, num_points);
    } else {
        tplanes_gather_direct_kernel<<<gather_blocks, 256, 0, stream>>>(coords, tp, out, num_points);
    }
}